// DistSAGE_13022340841577
// MI455X (gfx1250) — compile-verified
//
#include <hip/hip_runtime.h>
#include <hip/hip_bf16.h>
#include <math.h>

typedef float v2f __attribute__((ext_vector_type(2)));
typedef float v8f __attribute__((ext_vector_type(8)));

#define GNN_N 100000
#define D_IN 128
#define D_OUT 64

// ---------------------------------------------------------------- zero
__global__ void zero_f(float* __restrict__ p, size_t n) {
    size_t i = (size_t)blockIdx.x * blockDim.x + threadIdx.x;
    size_t stride = (size_t)gridDim.x * blockDim.x;
    for (; i < n; i += stride) p[i] = 0.0f;
}

// ---------------------------------------------------------------- degree
__global__ void deg_count(const int* __restrict__ dst, float* __restrict__ deg, int E) {
    int i = blockIdx.x * blockDim.x + threadIdx.x;
    if (i < E) atomicAdd(deg + dst[i], 1.0f);
}

// ---------------------------------------------------------------- scatter-add (gather src row, atomic-add onto dst row)
// One wave (32 lanes) per edge, float4 per lane -> 128 floats. L2-resident.
__global__ __launch_bounds__(256)
void scatter_add(const float* __restrict__ x, const int* __restrict__ src,
                 const int* __restrict__ dst, float* __restrict__ agg, int E) {
    long g = (long)blockIdx.x * blockDim.x + threadIdx.x;
    long edge = g >> 5;
    if (edge >= E) return;
    int lane = (int)(g & 31);
    long so = (long)src[edge] * D_IN + lane * 4;
    long dro = (long)dst[edge] * D_IN + lane * 4;
    float4 v = *(const float4*)(x + so);
    atomicAdd(agg + dro + 0, v.x);
    atomicAdd(agg + dro + 1, v.y);
    atomicAdd(agg + dro + 2, v.z);
    atomicAdd(agg + dro + 3, v.w);
}

// ---------------------------------------------------------------- fused SAGE conv + epilogue
// h = A @ Wr + (Agg/deg) @ Wn + bias ; EPI 0: LayerNorm+ReLU, EPI 1: log_softmax
// Block = 16 rows; NT waves, wave w computes 16-col tile [16w,16w+16).
// K = 128 fixed. W (out width) = 16*NT.
template<int NT, int EPI>
__global__ __launch_bounds__(32 * NT)
void sage_gemm(const float* __restrict__ A,     // [N,128] self features
               const float* __restrict__ Agg,   // [N,128] aggregated sums
               const float* __restrict__ deg,   // [N]
               const float* __restrict__ Wr,    // [128, 16*NT] row-major
               const float* __restrict__ Wn,    // [128, 16*NT]
               const float* __restrict__ bias,  // [16*NT]
               const float* __restrict__ gamma, // [16*NT] (EPI==0)
               const float* __restrict__ beta,  // [16*NT] (EPI==0)
               float* __restrict__ out)         // [N, 16*NT]
{
    constexpr int W = 16 * NT;
    constexpr int K = 128;
    constexpr int LDA = K + 4;               // +4 pad: bank = (4*row + k) % 64, conflict-free
    __shared__ float As[16][LDA];
    __shared__ float Gs[16][LDA];
    __shared__ float Hs[16][W];
    __shared__ float inv[16];

    const int tid  = threadIdx.x;
    const int lane = tid & 31;
    const int wave = tid >> 5;
    const long rowBase = (long)blockIdx.x * 16;

    if (tid < 16) {
        float d = deg[rowBase + tid];
        inv[tid] = 1.0f / fmaxf(d, 1.0f);
    }
    __syncthreads();

    // Cooperative stage: 16x128 self rows + normalized mean rows
    const float* aSrc = A + rowBase * K;
    const float* gSrc = Agg + rowBase * K;
    for (int e = tid; e < 16 * K; e += 32 * NT) {
        int r = e >> 7, c = e & 127;
        As[r][c] = aSrc[e];
        Gs[r][c] = gSrc[e] * inv[r];
    }
    __syncthreads();

    // WMMA fragment indices (ISA 7.12.2, 32-bit A 16x4 layout)
    const int n0   = wave * 16;
    const int arow = lane & 15;
    const int koff = (lane >> 4) << 1;   // lanes 0-15: K=k0,k0+1 ; lanes 16-31: K=k0+2,k0+3
    const int bcol = lane & 15;

    v8f acc = {};
#pragma unroll 4
    for (int k0 = 0; k0 < K; k0 += 4) {
        v2f a0, a1, b0, b1;
        a0.x = As[arow][k0 + koff];
        a0.y = As[arow][k0 + koff + 1];
        a1.x = Gs[arow][k0 + koff];
        a1.y = Gs[arow][k0 + koff + 1];
        b0.x = Wr[(k0 + koff) * W + n0 + bcol];
        b0.y = Wr[(k0 + koff + 1) * W + n0 + bcol];
        b1.x = Wn[(k0 + koff) * W + n0 + bcol];
        b1.y = Wn[(k0 + koff + 1) * W + n0 + bcol];
        // h += x @ Wr   (fp32 WMMA, D = A*B + C)
        acc = __builtin_amdgcn_wmma_f32_16x16x4_f32(
            false, a0, false, b0, (short)0, acc, false, false);
        // h += mean_agg @ Wn
        acc = __builtin_amdgcn_wmma_f32_16x16x4_f32(
            false, a1, false, b1, (short)0, acc, false, false);
    }

    // Spill C tile (+bias) to LDS: VGPR v holds row v + 8*(lane>=16)
    {
        float bv = bias[n0 + bcol];
        int rbase = (lane >> 4) << 3;
#pragma unroll
        for (int v = 0; v < 8; ++v)
            Hs[rbase + v][n0 + bcol] = acc[v] + bv;
    }
    __syncthreads();

    if constexpr (EPI == 0) {
        // LayerNorm + ReLU, one wave per 16/NT rows, wave32 shuffle reduce
        for (int r = wave; r < 16; r += NT) {
            float s = 0.0f;
            for (int c = lane; c < W; c += 32) s += Hs[r][c];
#pragma unroll
            for (int off = 16; off; off >>= 1) s += __shfl_xor(s, off, 32);
            float mu = s * (1.0f / W);
            float vv = 0.0f;
            for (int c = lane; c < W; c += 32) {
                float d = Hs[r][c] - mu;
                vv += d * d;
            }
#pragma unroll
            for (int off = 16; off; off >>= 1) vv += __shfl_xor(vv, off, 32);
            float rs = rsqrtf(vv * (1.0f / W) + 1e-5f);
            for (int c = lane; c < W; c += 32) {
                float val = (Hs[r][c] - mu) * rs * gamma[c] + beta[c];
                out[(rowBase + r) * W + c] = fmaxf(val, 0.0f);
            }
        }
    } else {
        // log_softmax over W classes
        for (int r = wave; r < 16; r += NT) {
            float m = -INFINITY;
            for (int c = lane; c < W; c += 32) m = fmaxf(m, Hs[r][c]);
#pragma unroll
            for (int off = 16; off; off >>= 1) m = fmaxf(m, __shfl_xor(m, off, 32));
            float s = 0.0f;
            for (int c = lane; c < W; c += 32) s += expf(Hs[r][c] - m);
#pragma unroll
            for (int off = 16; off; off >>= 1) s += __shfl_xor(s, off, 32);
            float ls = logf(s);
            for (int c = lane; c < W; c += 32)
                out[(rowBase + r) * W + c] = Hs[r][c] - m - ls;
        }
    }
}

// ---------------------------------------------------------------- launcher
extern "C" void kernel_launch(void* const* d_in, const int* in_sizes, int n_in,
                              void* d_out, int out_size, void* d_ws, size_t ws_size,
                              hipStream_t stream) {
    const float* x    = (const float*)d_in[0];
    const int*   esrc = (const int*)d_in[1];
    const int*   edst = (const int*)d_in[2];
    const float* Wr0  = (const float*)d_in[3];
    const float* Wn0  = (const float*)d_in[4];
    const float* b0   = (const float*)d_in[5];
    const float* g0   = (const float*)d_in[6];
    const float* be0  = (const float*)d_in[7];
    const float* Wr1  = (const float*)d_in[8];
    const float* Wn1  = (const float*)d_in[9];
    const float* b1   = (const float*)d_in[10];
    const float* g1   = (const float*)d_in[11];
    const float* be1  = (const float*)d_in[12];
    const float* Wr2  = (const float*)d_in[13];
    const float* Wn2  = (const float*)d_in[14];
    const float* b2   = (const float*)d_in[15];
    float* out = (float*)d_out;

    const int N = GNN_N;
    const int E = in_sizes[1];

    // workspace: deg[N] | agg[N*128] | h0[N*128] | h1[N*128]
    float* deg = (float*)d_ws;
    float* agg = deg + N;
    float* h0  = agg + (size_t)N * D_IN;
    float* h1  = h0 + (size_t)N * D_IN;

    const int ZB = 4096, ZT = 256;
    const int degBlocks = (E + 255) / 256;
    const int scatBlocks = (int)(((long)E * 32 + 255) / 256);
    const int gemmBlocks = N / 16;   // 6250, exact

    // zero deg + agg (contiguous)
    zero_f<<<ZB, ZT, 0, stream>>>(deg, (size_t)N * (D_IN + 1));
    deg_count<<<degBlocks, 256, 0, stream>>>(edst, deg, E);

    // ---- layer 0
    scatter_add<<<scatBlocks, 256, 0, stream>>>(x, esrc, edst, agg, E);
    sage_gemm<8, 0><<<gemmBlocks, 256, 0, stream>>>(x, agg, deg, Wr0, Wn0, b0, g0, be0, h0);

    // ---- layer 1
    zero_f<<<ZB, ZT, 0, stream>>>(agg, (size_t)N * D_IN);
    scatter_add<<<scatBlocks, 256, 0, stream>>>(h0, esrc, edst, agg, E);
    sage_gemm<8, 0><<<gemmBlocks, 256, 0, stream>>>(h0, agg, deg, Wr1, Wn1, b1, g1, be1, h1);

    // ---- layer 2 + log_softmax
    zero_f<<<ZB, ZT, 0, stream>>>(agg, (size_t)N * D_IN);
    scatter_add<<<scatBlocks, 256, 0, stream>>>(h1, esrc, edst, agg, E);
    sage_gemm<4, 1><<<gemmBlocks, 128, 0, stream>>>(h1, agg, deg, Wr2, Wn2, b2, nullptr, nullptr, out);
}